// FRNN_32160715113017
// MI455X (gfx1250) — compile-verified
//
#include <hip/hip_runtime.h>
#include <hip/hip_bf16.h>

typedef __attribute__((ext_vector_type(2))) float v2f;
typedef __attribute__((ext_vector_type(8))) float v8f;

#define T_STEPS 512
#define BATCH   32
#define DIM     1024
#define HID     1024
#define NG4     4096          // 4*HID
#define KTOT    2048          // DIM+HID
#define NTILES  256           // NG4/16
#define KSTEPS  512           // KTOT/4
#define KSTEP_X 256           // DIM/4 : ksteps [0,256) = Wx rows, [256,512) = Wh rows

// D = A(16x4) * B(4x16) + C, full fp32 (V_WMMA_F32_16X16X4_F32)
__device__ __forceinline__ v8f wmma4(v2f a, v2f b, v8f c) {
  return __builtin_amdgcn_wmma_f32_16x16x4_f32(false, a, false, b, (short)0, c, false, false);
}

__device__ __forceinline__ float sigmoidf(float x) {
  return 1.0f / (1.0f + __expf(-x));
}

// ---------------------------------------------------------------------------
// Zero h0, h1, c  (3 * BATCH*HID floats at the start of ws)
__global__ void FRNN_init_state(float* ws) {
  int i = blockIdx.x * blockDim.x + threadIdx.x;
  if (i < 3 * BATCH * HID) ws[i] = 0.0f;
}

// ---------------------------------------------------------------------------
// Pack W [2048,4096] (row-major) into WMMA B-fragment order:
//   Wp[((ntile*KSTEPS + kstep)*32 + lane)*2 + j]
//     = W[kstep*4 + 2*(lane>>4) + j][ntile*16 + (lane&15)]
// After packing, each wave's B operand for one WMMA is one contiguous 256B line.
__global__ void FRNN_pack_w(const float* __restrict__ W, float* __restrict__ Wp) {
  int idx = blockIdx.x * blockDim.x + threadIdx.x;     // NTILES*KSTEPS*32 items
  if (idx >= NTILES * KSTEPS * 32) return;
  int lane  = idx & 31;
  int kstep = (idx >> 5) & (KSTEPS - 1);
  int ntile = idx >> 14;
  int col = ntile * 16 + (lane & 15);
  int row = kstep * 4 + 2 * (lane >> 4);
  Wp[(size_t)idx * 2 + 0] = W[(size_t)row * NG4 + col];
  Wp[(size_t)idx * 2 + 1] = W[(size_t)(row + 1) * NG4 + col];
}

// ---------------------------------------------------------------------------
// Phase 1: pre[t,b,:] = x[t,b,:] @ Wx + bias.  M = Mrows (chunk*BATCH), N = 4096, K = 1024.
// Block = 4 waves; wave w -> M-tile (blockIdx.x*4+w), N-tile blockIdx.y.
__global__ void FRNN_gemm_x(const float* __restrict__ X,    // [Mrows, DIM]
                            const float* __restrict__ Wp,
                            const float* __restrict__ bias, // [NG4]
                            float* __restrict__ pre,        // [Mrows, NG4]
                            int Mrows) {
  const int wave = threadIdx.x >> 5;
  const int lane = threadIdx.x & 31;
  const int mtile = blockIdx.x * 4 + wave;
  const int ntile = blockIdx.y;
  if (mtile * 16 >= Mrows) return;   // wave-uniform: EXEC stays all-ones

  const int koff = 2 * (lane >> 4);
  const int mrow = mtile * 16 + (lane & 15);
  const float* ap = X + (size_t)mrow * DIM + koff;
  const float* bp = Wp + (size_t)ntile * (KSTEPS * 64) + lane * 2;   // kstep 0 base (Wx half)

  v8f acc0 = {}; v8f acc1 = {};
  #pragma unroll 4
  for (int ks = 0; ks < KSTEP_X; ks += 2) {
    v2f a0 = *(const v2f*)(ap + (size_t)ks * 4);
    v2f b0 = *(const v2f*)(bp + (size_t)ks * 64);
    acc0 = wmma4(a0, b0, acc0);
    v2f a1 = *(const v2f*)(ap + (size_t)(ks + 1) * 4);
    v2f b1 = *(const v2f*)(bp + (size_t)(ks + 1) * 64);
    acc1 = wmma4(a1, b1, acc1);
  }
  acc0 += acc1;

  const int col = ntile * 16 + (lane & 15);
  const float bv = bias[col];
  #pragma unroll
  for (int r = 0; r < 8; ++r) {
    int m = mtile * 16 + r + 8 * (lane >> 4);   // C/D layout: VGPR r -> row r / r+8
    pre[(size_t)m * NG4 + col] = acc0[r] + bv;
  }
}

// ---------------------------------------------------------------------------
// Phase 2: one time step.  gates = pre_t + h_in @ Wh; LSTM cell update.
// Grid = 64 blocks x 128 threads. Block owns hidden-col tile nt (16 cols);
// wave w computes gate w's 32x16 tile (two 16x16 WMMA M-tiles, K=1024,
// 8 independent accumulator chains of depth 64). The 4 gate tiles meet in
// LDS; after the workgroup barrier the block applies the fused cell update.
__global__ void FRNN_lstm_step(const float* __restrict__ hin,   // [32, HID]
                               float* __restrict__ hout,        // [32, HID]
                               float* __restrict__ c,           // [32, HID]
                               const float* __restrict__ Wp,
                               const float* __restrict__ pre_t) // [32, NG4]
{
  __shared__ float gsm[4 * BATCH * 16];          // [gate][b*16 + ci]
  const int wave = threadIdx.x >> 5;             // gate id 0..3
  const int lane = threadIdx.x & 31;
  const int nt   = blockIdx.x;                   // 0..63 hidden col tile
  const int koff = 2 * (lane >> 4);
  const int mrow = lane & 15;

  const float* a0p = hin + (size_t)mrow * HID + koff;          // rows 0..15
  const float* a1p = hin + (size_t)(mrow + 16) * HID + koff;   // rows 16..31
  // gate `wave` uses packed n-tile (wave*64 + nt), kstep range [256,512) (Wh half)
  const float* bp = Wp + ((size_t)(wave * 64 + nt) * KSTEPS + KSTEP_X) * 64 + lane * 2;

  v8f acc0[4] = {{}, {}, {}, {}};
  v8f acc1[4] = {{}, {}, {}, {}};
  for (int ks = 0; ks < 256; ks += 4) {
    __builtin_prefetch(bp + (size_t)(ks + 8) * 64, 0, 0);  // pull next Wh lines from L2
    #pragma unroll
    for (int q = 0; q < 4; ++q) {
      v2f a0 = *(const v2f*)(a0p + (size_t)(ks + q) * 4);
      v2f a1 = *(const v2f*)(a1p + (size_t)(ks + q) * 4);
      v2f b  = *(const v2f*)(bp  + (size_t)(ks + q) * 64);
      acc0[q] = wmma4(a0, b, acc0[q]);
      acc1[q] = wmma4(a1, b, acc1[q]);
    }
  }
  v8f s0 = (acc0[0] + acc0[1]) + (acc0[2] + acc0[3]);
  v8f s1 = (acc1[0] + acc1[1]) + (acc1[2] + acc1[3]);

  // Publish this gate's 32x16 tile to LDS (C/D layout: VGPR r -> row r / r+8)
  {
    const int ci = lane & 15;
    const int rb = 8 * (lane >> 4);
    #pragma unroll
    for (int r = 0; r < 8; ++r) {
      gsm[wave * 512 + (r + rb) * 16 + ci]      = s0[r];
      gsm[wave * 512 + (16 + r + rb) * 16 + ci] = s1[r];
    }
  }
  __syncthreads();

  // Fused LSTM cell update: 512 elements (32 rows x 16 cols) over 128 threads.
  const int colbase = nt * 16;
  #pragma unroll
  for (int k = 0; k < 4; ++k) {
    int e  = threadIdx.x + 128 * k;       // 0..511
    int b  = e >> 4;
    int ci = e & 15;
    int col = colbase + ci;
    const float* pr = pre_t + (size_t)b * NG4 + col;
    float gi = gsm[0 * 512 + e] + pr[0 * HID];
    float gj = gsm[1 * 512 + e] + pr[1 * HID];
    float gf = gsm[2 * 512 + e] + pr[2 * HID];
    float go = gsm[3 * 512 + e] + pr[3 * HID];
    size_t si = (size_t)b * HID + col;
    float cn = c[si] * sigmoidf(gf + 1.0f) + sigmoidf(gi) * tanhf(gj);
    c[si]    = cn;
    hout[si] = tanhf(cn) * sigmoidf(go);
  }
}

// ---------------------------------------------------------------------------
// d_out = concat(c_final, h_final) flat
__global__ void FRNN_copy_out(const float* __restrict__ c, const float* __restrict__ h,
                              float* __restrict__ out) {
  int i = blockIdx.x * blockDim.x + threadIdx.x;
  const int n = BATCH * HID;
  if (i < n)           out[i] = c[i];
  else if (i < 2 * n)  out[i] = h[i - n];
}

// ---------------------------------------------------------------------------
extern "C" void kernel_launch(void* const* d_in, const int* in_sizes, int n_in,
                              void* d_out, int out_size, void* d_ws, size_t ws_size,
                              hipStream_t stream) {
  const float* x    = (const float*)d_in[0];  // [512, 32, 1024]
  const float* W    = (const float*)d_in[1];  // [2048, 4096]
  const float* bias = (const float*)d_in[2];  // [4096]
  float* out = (float*)d_out;
  float* ws  = (float*)d_ws;

  float* h0 = ws;
  float* h1 = h0 + BATCH * HID;
  float* c  = h1 + BATCH * HID;
  float* Wp = c  + BATCH * HID;
  float* pre = Wp + (size_t)KTOT * NG4;

  // Adaptive chunking of the pre-activation buffer to whatever scratch we have.
  size_t fixed_floats = (size_t)3 * BATCH * HID + (size_t)KTOT * NG4;
  long long avail = (long long)(ws_size / 4) - (long long)fixed_floats;
  long long per_step = (long long)BATCH * NG4;
  int Tc = (avail > 0) ? (int)(avail / per_step) : 1;
  if (Tc < 1) Tc = 1;
  if (Tc > T_STEPS) Tc = T_STEPS;

  FRNN_init_state<<<(3 * BATCH * HID + 255) / 256, 256, 0, stream>>>(ws);
  FRNN_pack_w<<<(NTILES * KSTEPS * 32 + 255) / 256, 256, 0, stream>>>(W, Wp);

  int hb = 0;  // current h lives in (hb ? h1 : h0)
  for (int t0 = 0; t0 < T_STEPS; t0 += Tc) {
    int tc = T_STEPS - t0; if (tc > Tc) tc = Tc;
    int Mrows = tc * BATCH;
    dim3 grid1((unsigned)((Mrows / 16 + 3) / 4), NTILES);
    FRNN_gemm_x<<<grid1, 128, 0, stream>>>(x + (size_t)t0 * BATCH * DIM, Wp, bias, pre, Mrows);
    for (int t = 0; t < tc; ++t) {
      const float* hi = hb ? h1 : h0;
      float*       ho = hb ? h0 : h1;
      FRNN_lstm_step<<<64, 128, 0, stream>>>(hi, ho, c, Wp,
                                             pre + (size_t)t * BATCH * NG4);
      hb ^= 1;
    }
  }
  FRNN_copy_out<<<(2 * BATCH * HID + 255) / 256, 256, 0, stream>>>(
      c, hb ? h1 : h0, out);
}